// SGPR_Baseline_Attention_Fusion_62199716381016
// MI455X (gfx1250) — compile-verified
//
#include <hip/hip_runtime.h>
#include <math.h>

#define B_ 8
#define N_ 1024
#define KNN 20
#define NEGS 0.2f
#define BN_SCALE 0.9999950000374997f   // 1/sqrt(1+1e-5)

typedef float v2f __attribute__((ext_vector_type(2)));
typedef float v8f __attribute__((ext_vector_type(8)));

// ---------------------------------------------------------------------------
// xx[b,n] = sum_c x[b,c,n]^2   (x[b,c,n] = X[(b*Ctot+coff+c)*N + n])
// ---------------------------------------------------------------------------
__global__ void xsq_kernel(const float* __restrict__ X, int coff, int Ctot, int C,
                           float* __restrict__ xx) {
  int i = blockIdx.x * 256 + threadIdx.x;
  if (i >= B_ * N_) return;
  int b = i >> 10, n = i & 1023;
  float s = 0.f;
  for (int c = 0; c < C; ++c) {
    float v = X[(size_t)(b * Ctot + coff + c) * N_ + n];
    s = fmaf(v, v, s);
  }
  xx[i] = s;
}

// ---------------------------------------------------------------------------
// WMMA top-k (C compile-time): block = 16 rows (points) x 1024 cols of the
// neg-distance matrix. Phase 1: Gram tiles via V_WMMA_F32_16X16X4_F32 with
// A-fragments hoisted out of the column-tile loop; negdist into 64KB LDS.
// Phase 2: per-row top-20 via wave-level shuffle argmax (2 rows per wave).
// Only the SET of indices matters downstream (max over k is order-invariant).
// ---------------------------------------------------------------------------
template <int C>
__global__ void topk_kernel(const float* __restrict__ X, int coff, int Ctot,
                            const float* __restrict__ xx, int* __restrict__ idx) {
  constexpr int C4 = (C + 3) & ~3;
  extern __shared__ float dist[];  // [16][N_]
  const int b = blockIdx.y, n0 = blockIdx.x * 16, t = threadIdx.x;
  const int lane = t & 31, wave = t >> 5;
  const int half = lane >> 4, l = lane & 15;

  float xxr[8];
#pragma unroll
  for (int v = 0; v < 8; ++v) xxr[v] = xx[b * N_ + n0 + v + 8 * half];

  // A fragments: invariant across column tiles -> hoist into registers.
  v2f afrag[C4 / 4];
#pragma unroll
  for (int i = 0; i < C4 / 4; ++i) {
    const int k0 = i * 4 + 2 * half;
    afrag[i].x = (k0 < C)     ? X[(size_t)(b * Ctot + coff + k0) * N_ + n0 + l]     : 0.f;
    afrag[i].y = (k0 + 1 < C) ? X[(size_t)(b * Ctot + coff + k0 + 1) * N_ + n0 + l] : 0.f;
  }

  for (int ct = wave; ct < N_ / 16; ct += 8) {   // wave-uniform loop
    const int m0 = ct * 16;
    v8f acc = {0.f, 0.f, 0.f, 0.f, 0.f, 0.f, 0.f, 0.f};
#pragma unroll
    for (int i = 0; i < C4 / 4; ++i) {
      const int k0 = i * 4 + 2 * half;
      v2f bb;
      bb.x = (k0 < C)     ? X[(size_t)(b * Ctot + coff + k0) * N_ + m0 + l]     : 0.f;
      bb.y = (k0 + 1 < C) ? X[(size_t)(b * Ctot + coff + k0 + 1) * N_ + m0 + l] : 0.f;
      acc = __builtin_amdgcn_wmma_f32_16x16x4_f32(false, afrag[i], false, bb, (short)0,
                                                  acc, false, false);
    }
    const float xxc = xx[b * N_ + m0 + l];
#pragma unroll
    for (int v = 0; v < 8; ++v)
      dist[(v + 8 * half) * N_ + m0 + l] = 2.f * acc[v] - xxr[v] - xxc;
  }
  __syncthreads();

  for (int rs = 0; rs < 2; ++rs) {
    const int r = wave * 2 + rs;
    float* drow = dist + r * N_;
    const int gn = n0 + r;
    for (int s = 0; s < KNN; ++s) {
      float bv = -INFINITY; int bi = N_;
      for (int m = lane; m < N_; m += 32) {
        float v = drow[m];
        if (v > bv || (v == bv && m < bi)) { bv = v; bi = m; }
      }
#pragma unroll
      for (int off = 16; off > 0; off >>= 1) {
        float ov = __shfl_xor(bv, off, 32);
        int oi = __shfl_xor(bi, off, 32);
        if (ov > bv || (ov == bv && oi < bi)) { bv = ov; bi = oi; }
      }
      if (lane == 0) idx[((size_t)b * N_ + gn) * KNN + s] = bi;
      drow[bi] = -INFINITY;   // all lanes: same addr, same value (benign)
    }
  }
}

// ---------------------------------------------------------------------------
// WMMA edge layer (C, O compile-time; one block per point n):
//   y[o,k] = w1[o,:]·nb[:,k] + cen[o],  cen[o] = b[o] + (w2-w1)[o,:]·xn
//   z = leaky(y*g*BN_SCALE + beta), Y[o] = max_k z  (k>=20 masked to -inf)
// GEMM (O x C) @ (C x 32) via WMMA tiles; wave-uniform tile loop.
// ---------------------------------------------------------------------------
template <int C, int O>
__global__ void edge_kernel(const float* __restrict__ X, int coff, int Ctot,
                            const int* __restrict__ idx,
                            const float* __restrict__ W, const float* __restrict__ bias,
                            const float* __restrict__ gma, const float* __restrict__ bta,
                            float* __restrict__ Y) {
  constexpr int C4 = (C + 3) & ~3;
  const int b = blockIdx.y, n = blockIdx.x, t = threadIdx.x;
  const int lane = t & 31, wave = t >> 5;
  const int half = lane >> 4, l = lane & 15;
  __shared__ float xn[C];
  __shared__ float nbc[C4][32];   // [c][k], zero-padded cols 20..31 and c>=C
  __shared__ float cenl[O], scl[O], btl[O];
  __shared__ float ymax[O][2];
  __shared__ int jj[KNN];

  for (int c = t; c < C; c += 256) xn[c] = X[(size_t)(b * Ctot + coff + c) * N_ + n];
  if (t < KNN) jj[t] = idx[((size_t)b * N_ + n) * KNN + t];
  for (int e = t; e < C4 * 32; e += 256) nbc[e >> 5][e & 31] = 0.f;
  __syncthreads();
  for (int e = t; e < C * KNN; e += 256) {
    int c = e / KNN, k = e - c * KNN;
    nbc[c][k] = X[(size_t)(b * Ctot + coff + c) * N_ + jj[k]];
  }
  if (t < O) {
    const float* w1 = W + (size_t)t * 2 * C;
    const float* w2 = w1 + C;
    float cen = bias[t];
#pragma unroll
    for (int c = 0; c < C; ++c) cen = fmaf(w2[c] - w1[c], xn[c], cen);
    cenl[t] = cen;
    scl[t] = gma[t] * BN_SCALE;
    btl[t] = bta[t];
  }
  __syncthreads();

  constexpr int tiles = (O >> 4) * 2;
#pragma unroll
  for (int tile = wave; tile < tiles; tile += 8) {   // wave-uniform loop
    const int tm = tile >> 1, tn = tile & 1;
    const int arow = tm * 16 + l;
    v8f acc = {0.f, 0.f, 0.f, 0.f, 0.f, 0.f, 0.f, 0.f};
#pragma unroll
    for (int i = 0; i < C4 / 4; ++i) {
      const int k0 = i * 4 + 2 * half;
      v2f a, bb;
      a.x = (k0 < C)     ? W[(size_t)arow * 2 * C + k0]     : 0.f;
      a.y = (k0 + 1 < C) ? W[(size_t)arow * 2 * C + k0 + 1] : 0.f;
      bb.x = nbc[k0][tn * 16 + l];
      bb.y = nbc[k0 + 1][tn * 16 + l];
      acc = __builtin_amdgcn_wmma_f32_16x16x4_f32(false, a, false, bb, (short)0, acc,
                                                  false, false);
    }
#pragma unroll
    for (int v = 0; v < 8; ++v) {
      const int o = tm * 16 + v + 8 * half;
      const int k = tn * 16 + l;
      float z;
      if (k < KNN) {
        z = fmaf(acc[v] + cenl[o], scl[o], btl[o]);
        z = z >= 0.f ? z : NEGS * z;
      } else {
        z = -INFINITY;
      }
#pragma unroll
      for (int off = 8; off > 0; off >>= 1) z = fmaxf(z, __shfl_xor(z, off, 32));
      if (l == 0) ymax[o][tn] = z;
    }
  }
  __syncthreads();
  if (t < O) Y[((size_t)b * O + t) * N_ + n] = fmaxf(ymax[t][0], ymax[t][1]);
}

// ---------------------------------------------------------------------------
// WMMA f32 GEMM: C[M,N] = A[M,K] @ B(K,N) (+bias[n]). BT: B given as (N,K).
// ---------------------------------------------------------------------------
template <bool BT, bool BIAS>
__global__ void wmma_gemm_kernel(const float* __restrict__ A, const float* __restrict__ Bm,
                                 const float* __restrict__ bias, float* __restrict__ C,
                                 int M, int Nn, int Kk) {
  const int lane = threadIdx.x & 31;
  const int wave = threadIdx.x >> 5;
  const int tilesN = Nn >> 4;
  const int tile = blockIdx.x * 8 + wave;
  const int total = (M >> 4) * tilesN;
  if (tile >= total) return;                 // wave-uniform: EXEC stays all-1s
  const int tm = tile / tilesN, tn = tile % tilesN;
  const int half = lane >> 4, l = lane & 15;
  const int arow = tm * 16 + l;
  const int bcol = tn * 16 + l;
  v8f c = {0.f, 0.f, 0.f, 0.f, 0.f, 0.f, 0.f, 0.f};
  for (int kk = 0; kk < Kk; kk += 4) {
    const int k0 = kk + 2 * half;
    v2f a, bb;
    a.x = A[(size_t)arow * Kk + k0];
    a.y = A[(size_t)arow * Kk + k0 + 1];
    if (BT) {
      bb.x = Bm[(size_t)bcol * Kk + k0];
      bb.y = Bm[(size_t)bcol * Kk + k0 + 1];
    } else {
      bb.x = Bm[(size_t)k0 * Nn + bcol];
      bb.y = Bm[(size_t)(k0 + 1) * Nn + bcol];
    }
    c = __builtin_amdgcn_wmma_f32_16x16x4_f32(false, a, false, bb, (short)0, c, false, false);
  }
  const float bv = BIAS ? bias[bcol] : 0.f;
  for (int v = 0; v < 8; ++v) {
    const int row = tm * 16 + v + 8 * half;
    C[(size_t)row * Nn + bcol] = c[v] + bv;
  }
}

// fuse[b,n,0:32]=xyz[b,f,n], fuse[b,n,32:64]=sem[b,f,n]
__global__ void fuse_kernel(const float* __restrict__ xyz, const float* __restrict__ sem,
                            float* __restrict__ fuse) {
  int i = blockIdx.x * 256 + threadIdx.x;
  if (i >= B_ * N_ * 64) return;
  int f = i & 63; int bn = i >> 6; int b = bn >> 10; int n = bn & 1023;
  fuse[i] = (f < 32) ? xyz[((size_t)b * 32 + f) * N_ + n]
                     : sem[((size_t)b * 32 + (f - 32)) * N_ + n];
}

// MHA core: per (b,n) block handles all 8 heads, d=8, softmax over 1024 keys.
__global__ void attn_kernel(const float* __restrict__ qkv, float* __restrict__ out) {
  const int b = blockIdx.y, n = blockIdx.x, t = threadIdx.x;
  __shared__ float q[64];
  __shared__ float srow[N_];
  __shared__ float red[256];
  __shared__ float racc[256 * 9];
  if (t < 64) q[t] = qkv[((size_t)b * N_ + n) * 192 + t];
  __syncthreads();
  const float scale = 0.35355339059327373f;  // 1/sqrt(8)
  for (int h = 0; h < 8; ++h) {
    float lmax = -INFINITY;
    for (int m = t; m < N_; m += 256) {
      const float* kp = &qkv[((size_t)b * N_ + m) * 192 + 64 + h * 8];
      float s = 0.f;
      for (int d = 0; d < 8; ++d) s = fmaf(q[h * 8 + d], kp[d], s);
      s *= scale;
      srow[m] = s;
      lmax = fmaxf(lmax, s);
    }
    red[t] = lmax;
    __syncthreads();
    for (int off = 128; off > 0; off >>= 1) {
      if (t < off) red[t] = fmaxf(red[t], red[t + off]);
      __syncthreads();
    }
    const float mx = red[0];
    float den = 0.f, acc[8] = {0.f, 0.f, 0.f, 0.f, 0.f, 0.f, 0.f, 0.f};
    for (int m = t; m < N_; m += 256) {
      float e = __expf(srow[m] - mx);
      den += e;
      const float* vp = &qkv[((size_t)b * N_ + m) * 192 + 128 + h * 8];
      for (int d = 0; d < 8; ++d) acc[d] = fmaf(e, vp[d], acc[d]);
    }
    racc[t * 9] = den;
    for (int d = 0; d < 8; ++d) racc[t * 9 + 1 + d] = acc[d];
    __syncthreads();
    for (int off = 128; off > 0; off >>= 1) {
      if (t < off)
        for (int j = 0; j < 9; ++j) racc[t * 9 + j] += racc[(t + off) * 9 + j];
      __syncthreads();
    }
    if (t < 8) out[((size_t)b * N_ + n) * 64 + h * 8 + t] = racc[1 + t] / racc[0];
    __syncthreads();
  }
}

// mean of adjacent channel pairs: (B,N,64) -> (B,N,32)
__global__ void pairmean_kernel(const float* __restrict__ x, float* __restrict__ emb) {
  int i = blockIdx.x * 256 + threadIdx.x;
  if (i >= B_ * N_ * 32) return;
  int f = i & 31; int bn = i >> 5;
  emb[i] = 0.5f * (x[(size_t)bn * 64 + 2 * f] + x[(size_t)bn * 64 + 2 * f + 1]);
}

// g[b,f] = tanh(mean_n tmp[b,n,f])
__global__ void meantanh_kernel(const float* __restrict__ tmp, float* __restrict__ g) {
  int b = blockIdx.x, t = threadIdx.x;
  int f = t & 31, grp = t >> 5;
  __shared__ float red[256];
  float s = 0.f;
  for (int n = grp; n < N_; n += 8) s += tmp[((size_t)b * N_ + n) * 32 + f];
  red[t] = s;
  __syncthreads();
  for (int off = 4; off > 0; off >>= 1) {
    if (grp < off) red[t] += red[t + off * 32];
    __syncthreads();
  }
  if (t < 32) g[b * 32 + t] = tanhf(red[t] * (1.0f / N_));
}

// s[b,n] = sigmoid(emb[b,n,:]·g[b,:])
__global__ void score_kernel(const float* __restrict__ emb, const float* __restrict__ g,
                             float* __restrict__ sout) {
  int i = blockIdx.x * 256 + threadIdx.x;
  if (i >= B_ * N_) return;
  int b = i >> 10;
  float s = 0.f;
  for (int f = 0; f < 32; ++f) s = fmaf(emb[(size_t)i * 32 + f], g[b * 32 + f], s);
  sout[i] = 1.f / (1.f + __expf(-s));
}

// p[b,f] = sum_n emb[b,n,f] * s[b,n]
__global__ void pool_kernel(const float* __restrict__ emb, const float* __restrict__ s,
                            float* __restrict__ p) {
  int b = blockIdx.x, t = threadIdx.x;
  int f = t & 31, grp = t >> 5;
  __shared__ float red[256];
  float acc = 0.f;
  for (int n = grp; n < N_; n += 8) acc = fmaf(emb[((size_t)b * N_ + n) * 32 + f], s[b * N_ + n], acc);
  red[t] = acc;
  __syncthreads();
  for (int off = 4; off > 0; off >>= 1) {
    if (grp < off) red[t] += red[t + off * 32];
    __syncthreads();
  }
  if (t < 32) p[b * 32 + t] = red[t];
}

// NTN + 2-layer MLP + sigmoid. Single block.
__global__ void final_kernel(const float* __restrict__ p1, const float* __restrict__ p2,
                             const float* __restrict__ ntn_w, const float* __restrict__ ntn_v,
                             const float* __restrict__ ntn_b, const float* __restrict__ fc1_w,
                             const float* __restrict__ fc1_b, const float* __restrict__ sc_w,
                             const float* __restrict__ sc_b, float* __restrict__ score) {
  __shared__ float dd[B_][32];
  __shared__ float tt[B_][16];
  __shared__ float hh[B_][16];
  int t = threadIdx.x;
  if (t < B_ * 32) {
    int b = t >> 5, f = t & 31;
    dd[b][f] = p1[b * 32 + f] - p2[b * 32 + f];
  }
  __syncthreads();
  if (t < B_ * 16) {
    int b = t >> 4, j = t & 15;
    float acc = ntn_b[j];
    for (int f = 0; f < 32; ++f) {
      acc = fmaf(dd[b][f], ntn_v[j * 32 + f], acc);
      float inner = 0.f;
      for (int gg = 0; gg < 32; ++gg)
        inner = fmaf(ntn_w[((size_t)f * 32 + gg) * 16 + j], dd[b][gg], inner);
      acc = fmaf(dd[b][f], inner, acc);
    }
    tt[b][j] = fmaxf(acc, 0.f);
  }
  __syncthreads();
  if (t < B_ * 16) {
    int b = t >> 4, j = t & 15;
    float h = fc1_b[j];
    for (int i = 0; i < 16; ++i) h = fmaf(fc1_w[j * 16 + i], tt[b][i], h);
    hh[b][j] = fmaxf(h, 0.f);
  }
  __syncthreads();
  if (t < B_) {
    float sc = sc_b[0];
    for (int i = 0; i < 16; ++i) sc = fmaf(sc_w[i], hh[t][i], sc);
    score[t] = 1.f / (1.f + __expf(-sc));
  }
}

// ---------------------------------------------------------------------------
static void run_gemm(const float* A, const float* Bm, const float* bias, float* C,
                     int M, int Nn, int Kk, bool bt, bool hasbias, hipStream_t stream) {
  int tiles = (M / 16) * (Nn / 16);
  int blocks = (tiles + 7) / 8;
  if (bt) {
    if (hasbias) wmma_gemm_kernel<true, true><<<blocks, 256, 0, stream>>>(A, Bm, bias, C, M, Nn, Kk);
    else         wmma_gemm_kernel<true, false><<<blocks, 256, 0, stream>>>(A, Bm, bias, C, M, Nn, Kk);
  } else {
    if (hasbias) wmma_gemm_kernel<false, true><<<blocks, 256, 0, stream>>>(A, Bm, bias, C, M, Nn, Kk);
    else         wmma_gemm_kernel<false, false><<<blocks, 256, 0, stream>>>(A, Bm, bias, C, M, Nn, Kk);
  }
}

static void run_knn(const float* X, int coff, int Ctot, int C, float* xxbuf, int* idxbuf,
                    hipStream_t stream) {
  xsq_kernel<<<(B_ * N_ + 255) / 256, 256, 0, stream>>>(X, coff, Ctot, C, xxbuf);
  const dim3 g(N_ / 16, B_);
  const size_t sh = 16 * N_ * sizeof(float);
  switch (C) {
    case 3:   topk_kernel<3>  <<<g, 256, sh, stream>>>(X, coff, Ctot, xxbuf, idxbuf); break;
    case 12:  topk_kernel<12> <<<g, 256, sh, stream>>>(X, coff, Ctot, xxbuf, idxbuf); break;
    case 64:  topk_kernel<64> <<<g, 256, sh, stream>>>(X, coff, Ctot, xxbuf, idxbuf); break;
    default:  topk_kernel<128><<<g, 256, sh, stream>>>(X, coff, Ctot, xxbuf, idxbuf); break;
  }
}

extern "C" void kernel_launch(void* const* d_in, const int* in_sizes, int n_in,
                              void* d_out_v, int out_size, void* d_ws, size_t ws_size,
                              hipStream_t stream) {
  const float* feats[2] = {(const float*)d_in[0], (const float*)d_in[1]};
  const float* cw[3] = {(const float*)d_in[2], (const float*)d_in[6], (const float*)d_in[10]};
  const float* cb[3] = {(const float*)d_in[3], (const float*)d_in[7], (const float*)d_in[11]};
  const float* cg[3] = {(const float*)d_in[4], (const float*)d_in[8], (const float*)d_in[12]};
  const float* cB[3] = {(const float*)d_in[5], (const float*)d_in[9], (const float*)d_in[13]};
  const float* sw[3] = {(const float*)d_in[14], (const float*)d_in[18], (const float*)d_in[22]};
  const float* sb[3] = {(const float*)d_in[15], (const float*)d_in[19], (const float*)d_in[23]};
  const float* sg[3] = {(const float*)d_in[16], (const float*)d_in[20], (const float*)d_in[24]};
  const float* sB[3] = {(const float*)d_in[17], (const float*)d_in[21], (const float*)d_in[25]};
  const float* mha_wi = (const float*)d_in[26];
  const float* mha_bi = (const float*)d_in[27];
  const float* mha_wo = (const float*)d_in[28];
  const float* mha_bo = (const float*)d_in[29];
  const float* att_w = (const float*)d_in[30];
  const float* ntn_w = (const float*)d_in[31];
  const float* ntn_v = (const float*)d_in[32];
  const float* ntn_b = (const float*)d_in[33];
  const float* fc1_w = (const float*)d_in[34];
  const float* fc1_b = (const float*)d_in[35];
  const float* sc_w = (const float*)d_in[36];
  const float* sc_b = (const float*)d_in[37];
  float* out = (float*)d_out_v;

  char* wsp = (char*)d_ws;
  auto alloc = [&](size_t nbytes) -> void* {
    void* p = (void*)wsp;
    wsp += ((nbytes + 255) / 256) * 256;
    return p;
  };
  float* xA      = (float*)alloc((size_t)B_ * 128 * N_ * 4);
  float* xBuf    = (float*)alloc((size_t)B_ * 128 * N_ * 4);
  float* xyzout  = (float*)alloc((size_t)B_ * 32 * N_ * 4);
  float* semout  = (float*)alloc((size_t)B_ * 32 * N_ * 4);
  int*   idxbuf  = (int*)alloc((size_t)B_ * N_ * KNN * 4);
  float* xxbuf   = (float*)alloc((size_t)B_ * N_ * 4);
  float* fusebuf = (float*)alloc((size_t)B_ * N_ * 64 * 4);
  float* qkv     = (float*)alloc((size_t)B_ * N_ * 192 * 4);
  float* attnout = (float*)alloc((size_t)B_ * N_ * 64 * 4);
  float* mhaout  = (float*)alloc((size_t)B_ * N_ * 64 * 4);
  float* emb[2]  = {(float*)alloc((size_t)B_ * N_ * 32 * 4),
                    (float*)alloc((size_t)B_ * N_ * 32 * 4)};
  float* tmp32   = (float*)alloc((size_t)B_ * N_ * 32 * 4);
  float* gbuf    = (float*)alloc((size_t)B_ * 32 * 4);
  float* pbuf[2] = {(float*)alloc((size_t)B_ * 32 * 4), (float*)alloc((size_t)B_ * 32 * 4)};

  const dim3 gBN(N_, B_);
  const int M = B_ * N_;

  for (int which = 0; which < 2; ++which) {
    const float* f = feats[which];
    // xyz chain: (3->64), (64->128), (128->32)
    run_knn(f, 0, 15, 3, xxbuf, idxbuf, stream);
    edge_kernel<3, 64><<<gBN, 256, 0, stream>>>(f, 0, 15, idxbuf, cw[0], cb[0], cg[0], cB[0], xA);
    run_knn(xA, 0, 64, 64, xxbuf, idxbuf, stream);
    edge_kernel<64, 128><<<gBN, 256, 0, stream>>>(xA, 0, 64, idxbuf, cw[1], cb[1], cg[1], cB[1], xBuf);
    run_knn(xBuf, 0, 128, 128, xxbuf, idxbuf, stream);
    edge_kernel<128, 32><<<gBN, 256, 0, stream>>>(xBuf, 0, 128, idxbuf, cw[2], cb[2], cg[2], cB[2], xyzout);
    // sem chain: (12->64), (64->128), (128->32)
    run_knn(f, 3, 15, 12, xxbuf, idxbuf, stream);
    edge_kernel<12, 64><<<gBN, 256, 0, stream>>>(f, 3, 15, idxbuf, sw[0], sb[0], sg[0], sB[0], xA);
    run_knn(xA, 0, 64, 64, xxbuf, idxbuf, stream);
    edge_kernel<64, 128><<<gBN, 256, 0, stream>>>(xA, 0, 64, idxbuf, sw[1], sb[1], sg[1], sB[1], xBuf);
    run_knn(xBuf, 0, 128, 128, xxbuf, idxbuf, stream);
    edge_kernel<128, 32><<<gBN, 256, 0, stream>>>(xBuf, 0, 128, idxbuf, sw[2], sb[2], sg[2], sB[2], semout);
    // fuse + MHA
    fuse_kernel<<<(M * 64 + 255) / 256, 256, 0, stream>>>(xyzout, semout, fusebuf);
    run_gemm(fusebuf, mha_wi, mha_bi, qkv, M, 192, 64, true, true, stream);
    attn_kernel<<<gBN, 256, 0, stream>>>(qkv, attnout);
    run_gemm(attnout, mha_wo, mha_bo, mhaout, M, 64, 64, true, true, stream);
    pairmean_kernel<<<(M * 32 + 255) / 256, 256, 0, stream>>>(mhaout, emb[which]);
  }

  for (int which = 0; which < 2; ++which) {
    run_gemm(emb[which], att_w, nullptr, tmp32, M, 32, 32, false, false, stream);
    meantanh_kernel<<<B_, 256, 0, stream>>>(tmp32, gbuf);
    float* sdst = out + 8 + which * (B_ * N_);
    score_kernel<<<(M + 255) / 256, 256, 0, stream>>>(emb[which], gbuf, sdst);
    pool_kernel<<<B_, 256, 0, stream>>>(emb[which], sdst, pbuf[which]);
  }
  final_kernel<<<1, 256, 0, stream>>>(pbuf[0], pbuf[1], ntn_w, ntn_v, ntn_b,
                                      fc1_w, fc1_b, sc_w, sc_b, out);
}